// YoloLoss_57148834840778
// MI455X (gfx1250) — compile-verified
//
#include <hip/hip_runtime.h>

typedef __attribute__((ext_vector_type(2))) float v2f;
typedef __attribute__((ext_vector_type(8))) float v8f;

#define YS      14
#define CELL_C  30          // B*5 + NCLS = 30 floats per cell
#define SF_INV  (1.0f / 14.0f)
#define L_COORD 5.0f
#define L_NOOBJ 0.5f

// ds_swizzle_b32 offset encoding (group-of-32): [14:10]=xor, [9:5]=or, [4:0]=and
// SWAPX16: xor=0x10, or=0x00, and=0x1f  ->  0x401F  (swap lane halves, no LDS mem)
#define SWZ_SWAPX16 0x401F

// ---------------------------------------------------------------------------
// Wave32 sum via V_WMMA_F32_16X16X4_F32:
//   A (16x4 f32, 2 VGPRs): VGPR0 = acc (K=0 lanes 0-15, K=2 lanes 16-31),
//                          VGPR1 = 0   (K=1 / K=3)
//   B (4x16 f32) = all ones  =>  D[m][n] = acc[m] + acc[m+16]  (exact f32 adds)
//   Summing the 8 C/D VGPRs per lane: lanes 0-15 = sum rows 0..7,
//   lanes 16-31 = sum rows 8..15; one ds_swizzle SWAPX16 merges the halves.
// EXEC must be all ones: called uniformly from full 256-thread blocks.
// ---------------------------------------------------------------------------
__device__ __forceinline__ float wave_sum_wmma(float acc) {
    v2f a; a[0] = acc;  a[1] = 0.0f;
    v2f b; b[0] = 1.0f; b[1] = 1.0f;
    v8f c = {};
    c = __builtin_amdgcn_wmma_f32_16x16x4_f32(false, a, false, b,
                                              (short)0, c, false, false);
    float s = ((c[0] + c[1]) + (c[2] + c[3])) + ((c[4] + c[5]) + (c[6] + c[7]));
    int other = __builtin_amdgcn_ds_swizzle(__float_as_int(s), SWZ_SWAPX16);
    s += __int_as_float(other);   // every lane now holds the wave total
    return s;
}

// Deterministic block (256-thread, 8-wave) reduction. Result valid on tid 0.
__device__ __forceinline__ float block_sum(float v) {
    __shared__ float wsum[8];
    const int lane = threadIdx.x & 31;
    const int wave = threadIdx.x >> 5;
    float w = wave_sum_wmma(v);
    if (lane == 0) wsum[wave] = w;
    __syncthreads();
    float total = 0.0f;
    if (threadIdx.x == 0) {
#pragma unroll
        for (int i = 0; i < 8; ++i) total += wsum[i];   // fixed order
    }
    __syncthreads();
    return total;
}

// ---------------------------------------------------------------------------
// Per-cell YOLO loss (weights folded in; final /N applied in pass 2).
// Cell stride is 120 B (8-byte aligned), so stream with non-temporal b64 loads:
// data is read exactly once and must not churn the 192 MB L2.
// ---------------------------------------------------------------------------
__device__ __forceinline__ float cell_loss(const float* __restrict__ p,
                                           const float* __restrict__ t) {
    float pv[CELL_C], tv[CELL_C];
    const v2f* p2 = reinterpret_cast<const v2f*>(p);   // 8B-aligned ext-vector
    const v2f* t2 = reinterpret_cast<const v2f*>(t);
#pragma unroll
    for (int i = 0; i < CELL_C / 2; ++i) {
        v2f a = __builtin_nontemporal_load(p2 + i);
        v2f b = __builtin_nontemporal_load(t2 + i);
        pv[2 * i] = a[0]; pv[2 * i + 1] = a[1];
        tv[2 * i] = b[0]; tv[2 * i + 1] = b[1];
    }

    // IoU of each pred box vs target box 0
    const float tx = tv[0] * SF_INV, ty = tv[1] * SF_INV;
    const float tw = tv[2], th = tv[3];
    const float tl0 = tx - 0.5f * tw, tl1 = ty - 0.5f * th;
    const float tr0 = tx + 0.5f * tw, tr1 = ty + 0.5f * th;
    const float area_t = tw * th;

    float iou[2];
#pragma unroll
    for (int j = 0; j < 2; ++j) {
        const float* pb = pv + 5 * j;
        const float px = pb[0] * SF_INV, py = pb[1] * SF_INV;
        const float pw = pb[2], ph = pb[3];
        const float pl0 = px - 0.5f * pw, pl1 = py - 0.5f * ph;
        const float pr0 = px + 0.5f * pw, pr1 = py + 0.5f * ph;
        const float l0 = fmaxf(pl0, tl0), l1 = fmaxf(pl1, tl1);
        const float r0 = fminf(pr0, tr0), r1 = fminf(pr1, tr1);
        const float w0 = fmaxf(r0 - l0, 0.0f), w1 = fmaxf(r1 - l1, 0.0f);
        const float inter = w0 * w1;
        const float area_p = pw * ph;
        iou[j] = inter / (area_p + area_t - inter);
    }
    const int   best    = (iou[1] > iou[0]) ? 1 : 0;  // argmax tie -> 0
    const float iou_max = fmaxf(iou[0], iou[1]);

    const float* pr = pv + 5 * best;
    const float* tr = tv + 5 * best;
    const float dc  = iou_max - pr[4];
    const float contain = dc * dc;
    const float e0 = tr[0] - pr[0];
    const float e1 = tr[1] - pr[1];
    const float e2 = __builtin_sqrtf(tr[2]) - __builtin_sqrtf(pr[2]);
    const float e3 = __builtin_sqrtf(tr[3]) - __builtin_sqrtf(pr[3]);
    const float reg = e0 * e0 + e1 * e1 + e2 * e2 + e3 * e3;

    float cls = 0.0f;
#pragma unroll
    for (int c = 10; c < CELL_C; ++c) {
        const float d = tv[c] - pv[c];
        cls += d * d;
    }
    const float d4 = tv[4] - pv[4];
    const float d9 = tv[9] - pv[9];
    const float noobj = d4 * d4 + d9 * d9;

    const float objf   = (tv[4] > 0.0f)  ? 1.0f : 0.0f;
    const float noobjf = (tv[4] == 0.0f) ? 1.0f : 0.0f;

    return L_NOOBJ * noobjf * noobj + objf * (cls + contain + L_COORD * reg);
}

// Pass 1: grid-stride over cells, one f32 partial per block (deterministic).
__global__ void __launch_bounds__(256)
yolo_partial(const float* __restrict__ pred, const float* __restrict__ target,
             float* __restrict__ partials, long long ncells) {
    const long long idx0   = (long long)blockIdx.x * blockDim.x + threadIdx.x;
    const long long stride = (long long)gridDim.x * blockDim.x;
    float acc = 0.0f;
    for (long long cell = idx0; cell < ncells; cell += stride) {
        const float* p = pred   + cell * CELL_C;
        const float* t = target + cell * CELL_C;
        if (cell + stride < ncells) {                 // global_prefetch_b8 into GL2
            __builtin_prefetch(p + stride * CELL_C, 0, 0);
            __builtin_prefetch(t + stride * CELL_C, 0, 0);
        }
        acc += cell_loss(p, t);
    }
    const float bsum = block_sum(acc);                // WMMA wave reduce inside
    if (threadIdx.x == 0) partials[blockIdx.x] = bsum;
}

// Pass 2: single block folds the block partials and scales by 1/N.
__global__ void __launch_bounds__(256)
yolo_final(const float* __restrict__ partials, int n,
           float* __restrict__ out, float invN) {
    float acc = 0.0f;
    for (int i = threadIdx.x; i < n; i += blockDim.x) acc += partials[i];
    const float total = block_sum(acc);
    if (threadIdx.x == 0) out[0] = total * invN;
}

extern "C" void kernel_launch(void* const* d_in, const int* in_sizes, int n_in,
                              void* d_out, int out_size, void* d_ws, size_t ws_size,
                              hipStream_t stream) {
    (void)n_in; (void)out_size;
    const float* pred   = (const float*)d_in[0];
    const float* target = (const float*)d_in[1];
    float*       out    = (float*)d_out;
    float*       parts  = (float*)d_ws;

    const long long nelem  = (long long)in_sizes[0];
    const long long ncells = nelem / CELL_C;               // 1,605,632
    const long long nimg   = ncells / (YS * YS);           // 8192

    const int threads = 256;
    // ~2 cells per thread keeps plenty of waves in flight for 23.3 TB/s streaming.
    long long blocks = (ncells + (long long)threads * 2 - 1) / ((long long)threads * 2);
    const long long ws_slots = (long long)(ws_size / sizeof(float));
    if (ws_slots > 0 && blocks > ws_slots) blocks = ws_slots;  // grid-stride stays correct
    if (blocks < 1) blocks = 1;

    yolo_partial<<<(unsigned)blocks, threads, 0, stream>>>(pred, target, parts, ncells);
    yolo_final<<<1, threads, 0, stream>>>(parts, (int)blocks, out,
                                          1.0f / (float)nimg);
}